// FraudDetectionHybrid_65481071400316
// MI455X (gfx1250) — compile-verified
//
#include <hip/hip_runtime.h>

// FraudDetectionHybrid (quantum-gated LSTM) for MI455X / gfx1250.
//
// Exact collapse: gates broadcast over H => hidden state is a scalar per batch
// element; output reads only batch index B-1. We run the scalar recurrence for
// batch 2047 only. The x-dependent pre-activations are a 512x2 @ 2x16 GEMM,
// computed with V_WMMA_F32_16X16X4_F32 (K=2 padded to 4 with zeros).
// Kernel is latency-bound on the 512-step serial recurrence; phase 2 is
// instruction-count-minimized (v_cos/v_exp/v_rcp + lane-parallel gates).

typedef float v2f __attribute__((ext_vector_type(2)));
typedef float v4f __attribute__((ext_vector_type(4)));
typedef float v8f __attribute__((ext_vector_type(8)));

#define TT    512
#define BATCH 2048
#define DIMX  2
#define HDIM  64
#define CSTR  516   // column stride for apre: 16B-aligned, banks=(4n+t)%64 -> conflict-free

__device__ __forceinline__ const float* gate_ptr(int G, const float* f, const float* i,
                                                 const float* u, const float* o) {
  return (G == 0) ? f : (G == 1) ? i : (G == 2) ? u : o;
}

__device__ __forceinline__ float fast_sigm(float x) {
  // 1/(1+e^-x) with v_exp_f32 + v_rcp_f32
  return __builtin_amdgcn_rcpf(1.f + __expf(-x));
}

__global__ __launch_bounds__(256) void qlstm_kernel(
    const float* __restrict__ seq,
    const float* __restrict__ Wf, const float* __restrict__ bf, const float* __restrict__ pf,
    const float* __restrict__ Wi, const float* __restrict__ bi, const float* __restrict__ pi,
    const float* __restrict__ Wu, const float* __restrict__ bu, const float* __restrict__ pu,
    const float* __restrict__ Wo, const float* __restrict__ bo, const float* __restrict__ po,
    const float* __restrict__ Wc, const float* __restrict__ bc,
    float* __restrict__ out)
{
  __shared__ float apre[16 * CSTR];  // column-major: apre[n*CSTR + t], n = G*4+(q-1)
  __shared__ float hbuf[TT];         // h_t for batch B-1
  __shared__ float sbuf[17];         // 16 recurrent col-sums + sum(Wc)

  const int tid  = threadIdx.x;
  const int lane = tid & 31;
  const int wave = tid >> 5;

  // --- recurrent-weight column sums: S[G*4+k] = sum_{r=2..65} W_G[r][k+1]
  if (tid < 16) {
    const int G = tid >> 2, k = tid & 3;
    float s = 0.f;
    if (k < 3) {
      const float* W = gate_ptr(G, Wf, Wi, Wu, Wo);
      const int q = k + 1;
      for (int r = 2; r < 2 + HDIM; ++r) s += W[r * 4 + q];
    }
    sbuf[tid] = s;
  } else if (tid == 16) {
    float s = 0.f;
    for (int r = 0; r < HDIM; ++r) s += Wc[r];
    sbuf[16] = s;
  }

  // --- phase 1: A_pre = X(512x2) @ Wx(2x16) + (b+p) via v_wmma_f32_16x16x4_f32
  // A layout (ISA 7.12.2, 32-bit A 16x4): lanes 0-15 hold K=0 (V0), K=1 (V1);
  // lanes 16-31 hold K=2,3 -> zero padding.
  // B layout (K split across lane halves): lanes 0-15 = rows K=0 (V0), K=1 (V1);
  // lanes 16-31 = K=2,3 -> zero padding.
  const int  n16 = lane & 15;
  const bool hi  = lane >= 16;
  v2f  bmat = {0.f, 0.f};
  float cb  = 0.f;
  {
    const int G = n16 >> 2, k = n16 & 3;
    if (k < 3) {
      const int q = k + 1;
      const float* W  = gate_ptr(G, Wf, Wi, Wu, Wo);
      const float* bb = gate_ptr(G, bf, bi, bu, bo);
      const float* pp = gate_ptr(G, pf, pi, pu, po);
      if (!hi) { bmat[0] = W[q]; bmat[1] = W[4 + q]; }   // W[d=0][q], W[d=1][q]
      cb = bb[q] + pp[q];
    }
  }
  v8f cacc;
  #pragma unroll
  for (int v = 0; v < 8; ++v) cacc[v] = cb;   // C[m][n] = bias[n] for all m

  for (int j = 0; j < 4; ++j) {
    const int tbase = (wave * 4 + j) * 16;    // 8 waves x 4 tiles = 32 tiles of 16 rows
    v2f a = {0.f, 0.f};
    if (!hi) {
      const size_t idx = (((size_t)(tbase + lane)) * BATCH + (BATCH - 1)) * DIMX;
      a[0] = seq[idx];        // x0, K=0
      a[1] = seq[idx + 1];    // x1, K=1
    }
    v8f d = __builtin_amdgcn_wmma_f32_16x16x4_f32(false, a, false, bmat,
                                                  (short)0, cacc, false, false);
    // C/D layout: VGPR v = row v (+8 for hi lane half); column-major LDS makes
    // the 8 values contiguous -> two ds_store_b128.
    const int tb = tbase + (hi ? 8 : 0);
    v4f lo = {d[0], d[1], d[2], d[3]};
    v4f hi4 = {d[4], d[5], d[6], d[7]};
    *(v4f*)&apre[n16 * CSTR + tb]     = lo;
    *(v4f*)&apre[n16 * CSTR + tb + 4] = hi4;
  }
  __syncthreads();

  // --- phase 2: 512-step serial recurrence (wave 0; gates lane-parallel)
  if (wave == 0) {
    const float S    = sbuf[n16];
    const int   k    = lane & 3;
    const int   G    = (lane >> 2) & 3;
    const bool  live = (!hi) && (k < 3);
    const float kk   = (G == 2) ? 2.f : 1.f;  // tanh(x) = 2*sigm(2x)-1
    float h = 0.f, c = 0.f;
    for (int t4 = 0; t4 < TT; t4 += 4) {
      const v4f z4 = *(const v4f*)&apre[n16 * CSTR + t4];   // 4 steps of pre-acts
      v4f h4;
      #pragma unroll
      for (int u = 0; u < 4; ++u) {
        const float z  = z4[u] + h * S;
        float cv = live ? __cosf(z) : 1.f;
        cv *= __shfl_xor(cv, 1, 32);          // product within gate group of 4
        cv *= __shfl_xor(cv, 2, 32);          // lanes 4G..4G+3 hold gate-G product
        const float sg  = fast_sigm(kk * cv);
        const float act = (G == 2) ? (2.f * sg - 1.f) : sg;
        const float fg = __shfl(act, 0, 32);
        const float ig = __shfl(act, 4, 32);
        const float gg = __shfl(act, 8, 32);
        const float og = __shfl(act, 12, 32);
        c = fg * c + ig * gg;
        h = og * (2.f * fast_sigm(2.f * c) - 1.f);  // o * tanh(c)
        h4[u] = h;
      }
      *(v4f*)&hbuf[t4] = h4;  // all lanes write identical values (benign)
    }
  }
  __syncthreads();

  // --- phase 3: out[t] = sigmoid(h_t * sum(Wc) + bc)
  const float sWc = sbuf[16];
  const float bcv = bc[0];
  for (int t = tid; t < TT; t += 256) {
    out[t] = fast_sigm(hbuf[t] * sWc + bcv);
  }
}

extern "C" void kernel_launch(void* const* d_in, const int* in_sizes, int n_in,
                              void* d_out, int out_size, void* d_ws, size_t ws_size,
                              hipStream_t stream) {
  (void)in_sizes; (void)n_in; (void)out_size; (void)d_ws; (void)ws_size;
  const float* seq = (const float*)d_in[0];
  const float* Wf  = (const float*)d_in[1];
  const float* bf  = (const float*)d_in[2];
  const float* pf  = (const float*)d_in[3];
  const float* Wi  = (const float*)d_in[4];
  const float* bi  = (const float*)d_in[5];
  const float* pi  = (const float*)d_in[6];
  const float* Wu  = (const float*)d_in[7];
  const float* bu  = (const float*)d_in[8];
  const float* pu  = (const float*)d_in[9];
  const float* Wo  = (const float*)d_in[10];
  const float* bo  = (const float*)d_in[11];
  const float* po  = (const float*)d_in[12];
  const float* Wc  = (const float*)d_in[13];
  const float* bc  = (const float*)d_in[14];
  float* out = (float*)d_out;

  qlstm_kernel<<<1, 256, 0, stream>>>(seq, Wf, bf, pf, Wi, bi, pi,
                                      Wu, bu, pu, Wo, bo, po, Wc, bc, out);
}